// QuadraticDilation2D_489626272315
// MI455X (gfx1250) — compile-verified
//
#include <hip/hip_runtime.h>
#include <stdint.h>

typedef float v4f __attribute__((ext_vector_type(4)));

namespace {
constexpr int Bn = 4, Hn = 256, Wn = 256, Cn = 32, Fn = 8;
constexpr int WT = 8;                   // w-tile per block (one wave per w)
constexpr int HT = 32;                  // output rows per block
constexpr int SLOTS = WT + 4;           // 12 staged w positions (tile + 2-halo each side)
constexpr int ROWF = SLOTS * Cn;        // 384 floats per staged row
constexpr int ROWBYTES = ROWF * 4;      // 1536 bytes
constexpr int NBUF = 4;                 // 4 buffers -> 2-row-deep async pipeline
}

#define NINF (-__builtin_inff())

// CDNA5 async global->LDS copy (VGLOBAL opcode 98), tracked with ASYNCcnt.
// Operands: VDST = LDS byte address, VADDR = 64-bit global address, saddr=off.
__device__ __forceinline__ void async_load_b128(unsigned lds_off, const float* g) {
  asm volatile("global_load_async_to_lds_b128 %0, %1, off"
               :: "v"(lds_off), "v"(g) : "memory");
}

__global__ __launch_bounds__(256)
void quad_dilate_kernel(const float* __restrict__ x,
                        const float* __restrict__ bias,
                        float* __restrict__ out) {
  __shared__ __align__(16) float smem[NBUF * ROWF];   // 6 KB row stage

  const int tx   = threadIdx.x;
  const int wq   = tx >> 5;             // wave id = w within tile (0..7)
  const int lane = tx & 31;
  const int c0   = (lane & 7) * 4;      // 4 consecutive channels
  const int f0   = lane >> 3;           // 0..3 (thread also handles f0+4)

  const int w0 = blockIdx.x * WT;
  const int hs = blockIdx.y * HT;
  const int b  = blockIdx.z;
  const int w  = w0 + wq;

  // ---- async-loader role: 96 lanes move one 16B chunk each (12 slots x 128B)
  const int slot_a = tx >> 3;           // 0..11 for tx<96
  const int chunk  = tx & 7;            // 16B chunk within the 128B slot
  const int ws     = w0 - 2 + slot_a;   // global w of this slot
  const bool ws_ok = (ws >= 0) && (ws < Wn);
  const bool loader = (tx < 96) && ws_ok;

  const unsigned lds_base = (unsigned)(uintptr_t)(&smem[0]);
  const unsigned lds_my   = lds_base + (unsigned)((slot_a * Cn + chunk * 4) * 4);
  const float* gbase = x + (size_t)b * Hn * Wn * Cn + (ptrdiff_t)ws * Cn + chunk * 4;

  // ---- fill out-of-range W halo slots with -inf once (async loads skip them)
  if (tx < 96 && !ws_ok) {
    v4f ninf = {NINF, NINF, NINF, NINF};
#pragma unroll
    for (int q = 0; q < NBUF; ++q)
      *(v4f*)&smem[q * ROWF + slot_a * Cn + chunk * 4] = ninf;
  }

  // ---- per-(c,f) scaled bias:  q1 = -1/16 * b,  q2 = -4/16 * b
  float qb1[4][2], qb2[4][2];
#pragma unroll
  for (int j = 0; j < 4; ++j)
#pragma unroll
    for (int k = 0; k < 2; ++k) {
      float bv = bias[(c0 + j) * Fn + (f0 + 4 * k)];
      qb1[j][k] = -0.0625f * bv;
      qb2[j][k] = -0.25f   * bv;
    }

  float win[5][4][2];   // rolling W-dilated rows: [phase][channel][f-half]

  // ---- prologue: issue rows hs-2 (buf0) and hs-1 (buf1): 2-deep pipeline
  {
    const int r0 = hs - 2;
    if (loader && r0 >= 0)
      async_load_b128(lds_my, gbase + (size_t)r0 * (Wn * Cn));
    const int r1 = hs - 1;
    if (loader && r1 >= 0)
      async_load_b128(lds_my + (unsigned)ROWBYTES,
                      gbase + (size_t)r1 * (Wn * Cn));
  }

  float* op = out + (((size_t)b * Hn + hs) * Wn + w) * (Fn * Cn) + f0 * Cn + c0;

  int bcur = 0;                         // LDS buffer holding row for this step
  for (int ii = 0; ii < 8; ++ii) {      // 8 x 5 phases; steps >= 36 skipped
#pragma unroll
    for (int p = 0; p < 5; ++p) {
      const int i = ii * 5 + p;
      if (i < HT + 4) {                 // block-uniform guard: skip pad steps
        const int r = hs - 2 + i;       // row consumed this step

        // prefetch row r+2 (consumed 2 steps ahead); its target buffer was
        // last read at step i-2, fenced by the barrier at top of step i-1
        const int rn2 = r + 2;
        if (i < HT + 2 && loader && rn2 >= 0 && rn2 < Hn)
          async_load_b128(lds_my + (unsigned)(((bcur + 2) & 3) * ROWBYTES),
                          gbase + (size_t)rn2 * (Wn * Cn));

        // retire row r's loads (issued 2 steps ago); keep 2 newest in flight
        asm volatile("s_wait_asynccnt 2" ::: "memory");
        __syncthreads();

        // ---- W-pass: dilate row r along w into win[p]
        if (r >= 0 && r < Hn) {
          const float* bufp = &smem[bcur * ROWF];
          float xs[5][4];
#pragma unroll
          for (int d = 0; d < 5; ++d) {
            v4f q = *(const v4f*)&bufp[(wq + d) * Cn + c0];
            xs[d][0] = q[0]; xs[d][1] = q[1]; xs[d][2] = q[2]; xs[d][3] = q[3];
          }
#pragma unroll
          for (int j = 0; j < 4; ++j) {
            float m1 = fmaxf(xs[1][j], xs[3][j]);   // |dx|=1 pair (shared over f)
            float m2 = fmaxf(xs[0][j], xs[4][j]);   // |dx|=2 pair
#pragma unroll
            for (int k = 0; k < 2; ++k)
              win[p][j][k] = fmaxf(xs[2][j],
                                   fmaxf(m1 + qb1[j][k], m2 + qb2[j][k]));
          }
        } else {
#pragma unroll
          for (int j = 0; j < 4; ++j)
#pragma unroll
            for (int k = 0; k < 2; ++k) win[p][j][k] = NINF;
        }

        // ---- H-pass: combine 5 dilated rows -> output row ro = r-2
        if (i >= 4) {
          const int im2 = (p + 1) % 5;    // ro-2
          const int im1 = (p + 2) % 5;    // ro-1
          const int ic  = (p + 3) % 5;    // ro
          const int ip1 = (p + 4) % 5;    // ro+1
          const int ip2 = p;              // ro+2
          float res[2][4];
#pragma unroll
          for (int k = 0; k < 2; ++k)
#pragma unroll
            for (int j = 0; j < 4; ++j) {
              float a1 = fmaxf(win[im1][j][k], win[ip1][j][k]);
              float a2 = fmaxf(win[im2][j][k], win[ip2][j][k]);
              res[k][j] = fmaxf(win[ic][j][k],
                                fmaxf(a1 + qb1[j][k], a2 + qb2[j][k]));
            }
          v4f o0 = {res[0][0], res[0][1], res[0][2], res[0][3]};
          v4f o1 = {res[1][0], res[1][1], res[1][2], res[1][3]};
          __builtin_nontemporal_store(o0, (v4f*)op);              // f0 quad
          __builtin_nontemporal_store(o1, (v4f*)(op + 4 * Cn));   // f0+4 quad
          op += (size_t)Wn * (Fn * Cn);
        }
      }
      bcur = (bcur + 1) & 3;
    }
  }
}

extern "C" void kernel_launch(void* const* d_in, const int* in_sizes, int n_in,
                              void* d_out, int out_size, void* d_ws, size_t ws_size,
                              hipStream_t stream) {
  const float* x    = (const float*)d_in[0];
  const float* bias = (const float*)d_in[1];
  float* out        = (float*)d_out;
  dim3 grid(Wn / WT, Hn / HT, Bn);   // (32, 8, 4) blocks of 256 threads
  quad_dilate_kernel<<<grid, dim3(256), 0, stream>>>(x, bias, out);
}